// TemporalSpike_14396730376459
// MI455X (gfx1250) — compile-verified
//
#include <hip/hip_runtime.h>

typedef float v2f __attribute__((ext_vector_type(2)));
typedef float v8f __attribute__((ext_vector_type(8)));

namespace {
constexpr int kB = 16, kT = 128, kN = 512, kF = 32;
constexpr float kBeta = 0.9f, kThresh = 1.0f;
constexpr int kRowTiles = (kB * kN) / 16;  // 512 waves, 16 (b,n)-rows each
}

// One wave per 16 rows. State kept in registers across all 128 timesteps.
// D layout convention for state tiles: tile gt covers features gt*16..gt*16+15;
// VGPR j: lanes 0-15 = [f = gt*16 + j][row = lane], lanes 16-31 = [f = gt*16 + j + 8][row = lane-16].
__global__ __launch_bounds__(32)
void snn_rleaky_kernel(const float* __restrict__ x, const float* __restrict__ W,
                       const float* __restrict__ b_lin, float* __restrict__ out) {
  const int lane = threadIdx.x;   // 0..31
  const int tile = blockIdx.x;    // 0..511
  const int rlo  = lane & 15;     // row within tile / M-row for A
  const int hi   = lane >> 4;     // 0 = lanes 0-15, 1 = lanes 16-31

  const int b  = tile / (kN / 16);
  const int n0 = (tile % (kN / 16)) * 16;

  // --- W as A-matrix fragments (A[m=g][k=f] = W[g*32+f]); constant over time ---
  // 16x4 f32 A layout: lane m holds K = k0,k0+1 ; lane m+16 holds K = k0+2,k0+3.
  v2f Wa[2][8];
#pragma unroll
  for (int gt = 0; gt < 2; ++gt) {
#pragma unroll
    for (int s = 0; s < 8; ++s) {
      const int g = rlo + 16 * gt;
      const int f = s * 4 + hi * 2;
      Wa[gt][s].x = W[g * kF + f];
      Wa[gt][s].y = W[g * kF + f + 1];
    }
  }

  // --- bias broadcast in C/D layout: C[g][row] = b_lin[g] ---
  v8f Cinit[2];
#pragma unroll
  for (int gt = 0; gt < 2; ++gt)
#pragma unroll
    for (int j = 0; j < 8; ++j)
      Cinit[gt][j] = b_lin[gt * 16 + j + hi * 8];

  v8f M0 = {}, M1 = {};  // membrane potential, D layout
  v8f S0 = {}, S1 = {};  // spikes (exact 0/1), D layout

  const long long row_base = ((long long)b * kT) * kN + n0;   // in rows of kF floats
  const int lane_off = rlo * kF + hi * 8;                     // + gt*16 per tile
  float* const out_spk = out;
  float* const out_mem = out + (long long)kB * kT * kN * kF;

  for (int t = 0; t < kT; ++t) {
    const long long base = (row_base + (long long)t * kN) * kF + lane_off;
    // x_t fragments: per-lane 8 consecutive floats -> exactly the D layout.
    const v8f X0 = *(const v8f*)(x + base);        // f = 0..15 half-split
    const v8f X1 = *(const v8f*)(x + base + 16);   // f = 16..31

    // --- build B fragments (B[k=f][n=row] = spk_prev[f][row]) from D-layout spikes.
    // K-slice s covers f = 4s..4s+3. B 4x16 f32 layout: vgpr.x = K rows {k0 | k0+2},
    // vgpr.y = {k0+1 | k0+3}. Only half-wave swaps needed (shfl_xor 16 + select).
    v2f Bf[8];
#pragma unroll
    for (int s = 0; s < 8; ++s) {
      const v8f& Ss = (s < 4) ? S0 : S1;   // f 0..15 from S0, 16..31 from S1
      const int s2 = s & 3;
      const int j0 = (s2 & 1) * 4;         // source VGPR group within tile
      const float a0 = Ss[j0 + 0], a1 = Ss[j0 + 1];
      const float a2 = Ss[j0 + 2], a3 = Ss[j0 + 3];
      float bx, by;
      if (s2 < 2) {
        // f' = j0..j0+3 live in the LO halves of vj0..vj0+3
        const float w2 = __shfl_xor(a2, 16);
        const float w3 = __shfl_xor(a3, 16);
        bx = (lane < 16) ? a0 : w2;
        by = (lane < 16) ? a1 : w3;
      } else {
        // f' = j0+8..j0+11 live in the HI halves of vj0..vj0+3
        const float w0 = __shfl_xor(a0, 16);
        const float w1 = __shfl_xor(a1, 16);
        bx = (lane < 16) ? w0 : a2;
        by = (lane < 16) ? w1 : a3;
      }
      Bf[s].x = bx;
      Bf[s].y = by;
    }

    // --- recurrent matmul: rec[g][row] = sum_f W[g,f]*spk[f][row] + b_lin[g] ---
    v8f R0 = Cinit[0], R1 = Cinit[1];
#pragma unroll
    for (int s = 0; s < 8; ++s) {
      R0 = __builtin_amdgcn_wmma_f32_16x16x4_f32(false, Wa[0][s], false, Bf[s],
                                                 (short)0, R0, false, false);
      R1 = __builtin_amdgcn_wmma_f32_16x16x4_f32(false, Wa[1][s], false, Bf[s],
                                                 (short)0, R1, false, false);
    }

    // --- elementwise LIF update (subtract reset), then threshold ---
#pragma unroll
    for (int j = 0; j < 8; ++j) {
      const float m0 = kBeta * M0[j] + X0[j] + R0[j] - S0[j] * kThresh;
      const float m1 = kBeta * M1[j] + X1[j] + R1[j] - S1[j] * kThresh;
      M0[j] = m0;
      M1[j] = m1;
      S0[j] = (m0 - kThresh) > 0.0f ? 1.0f : 0.0f;
      S1[j] = (m1 - kThresh) > 0.0f ? 1.0f : 0.0f;
    }

    *(v8f*)(out_spk + base)      = S0;
    *(v8f*)(out_spk + base + 16) = S1;
    *(v8f*)(out_mem + base)      = M0;
    *(v8f*)(out_mem + base + 16) = M1;
  }
}

extern "C" void kernel_launch(void* const* d_in, const int* in_sizes, int n_in,
                              void* d_out, int out_size, void* d_ws, size_t ws_size,
                              hipStream_t stream) {
  const float* x     = (const float*)d_in[0];
  const float* W     = (const float*)d_in[1];
  const float* b_lin = (const float*)d_in[2];
  float* out = (float*)d_out;
  snn_rleaky_kernel<<<kRowTiles, 32, 0, stream>>>(x, W, b_lin, out);
}